// GTLayer_45621142618215
// MI455X (gfx1250) — compile-verified
//
#include <hip/hip_runtime.h>
#include <hip/hip_bf16.h>

typedef __attribute__((ext_vector_type(2))) float v2f;
typedef __attribute__((ext_vector_type(8))) float v8f;

#define LATDIM 128
#define NHEAD  4
#define DH     32

// ---------------------------------------------------------------------------
// Kernel 1: per-node QKV projection via V_WMMA_F32_16X16X4_F32.
// One wave (32 lanes) computes one 16x16 f32 output tile.
// grid.x = n_nodes/16 tiles of rows, grid.y = 24 (3 matrices * 8 column tiles)
//
// A (16x4 f32) VGPR layout (ISA 7.12.2): lanes 0-15 hold M=lane, K={0,1} in
// vgpr {0,1}; lanes 16-31 hold M=lane-16, K={2,3}.
// B (4x16) mirrored: vgpr i = W[K = k0 + i + 2*(lane>=16)][N = nbase + lane&15].
// C/D (16x16 f32): vgpr r = row (mbase + r + 8*(lane>=16)), col = lane&15.
// ---------------------------------------------------------------------------
__global__ __launch_bounds__(32)
void qkv_wmma_kernel(const float* __restrict__ embeds,
                     const float* __restrict__ qW,
                     const float* __restrict__ kW,
                     const float* __restrict__ vW,
                     float* __restrict__ Qo,
                     float* __restrict__ Ko,
                     float* __restrict__ Vo) {
    const int lane  = threadIdx.x;            // 0..31, wave32
    const int mbase = blockIdx.x * 16;
    const int which = blockIdx.y >> 3;        // 0=Q, 1=K, 2=V
    const int nbase = (blockIdx.y & 7) * 16;

    const float* __restrict__ W   = (which == 0) ? qW : (which == 1) ? kW : vW;
    float* __restrict__       Out = (which == 0) ? Qo : (which == 1) ? Ko : Vo;

    const int mrow  = mbase + (lane & 15);
    const int khalf = (lane >> 4) * 2;        // 0 for lanes 0-15, 2 for 16-31
    const int ncol  = nbase + (lane & 15);

    v8f acc = {};
    const float* __restrict__ arow = embeds + (size_t)mrow * LATDIM;

    #pragma unroll 4
    for (int k0 = 0; k0 < LATDIM; k0 += 4) {
        // A fragment: two consecutive K values per lane (8B aligned load)
        v2f a = *(const v2f*)(arow + k0 + khalf);
        // B fragment: two rows of W at this lane's column
        v2f b;
        b.x = W[(size_t)(k0 + khalf + 0) * LATDIM + ncol];
        b.y = W[(size_t)(k0 + khalf + 1) * LATDIM + ncol];
        acc = __builtin_amdgcn_wmma_f32_16x16x4_f32(
            /*neg_a=*/false, a, /*neg_b=*/false, b,
            /*c_mod=*/(short)0, acc, /*reuse_a=*/false, /*reuse_b=*/false);
    }

    const int cm = mbase + (lane >> 4) * 8;   // first output row for this lane
    #pragma unroll
    for (int r = 0; r < 8; ++r)
        Out[(size_t)(cm + r) * LATDIM + ncol] = acc[r];
}

// ---------------------------------------------------------------------------
// Kernel 2: per-edge attention logits. One wave per edge.
// Lane l handles dims [4l, 4l+4); head h = l/8 (dims h*32..h*32+31).
// Shuffle-reduce the 8 lanes of each head, then clip/exp, store expAtt and
// atomically accumulate the per-(row,head) softmax denominator.
// ---------------------------------------------------------------------------
__global__ __launch_bounds__(256)
void edge_logits_kernel(const float* __restrict__ Q,
                        const float* __restrict__ K,
                        const int*   __restrict__ rows,
                        const int*   __restrict__ cols,
                        float* __restrict__ expAtt,   // [E, NHEAD]
                        float* __restrict__ attNorm,  // [N, NHEAD]
                        int n_edges) {
    const int gid  = blockIdx.x * blockDim.x + threadIdx.x;
    const int e    = gid >> 5;
    const int lane = threadIdx.x & 31;
    if (e >= n_edges) return;

    const int row = rows[e];
    const int col = cols[e];

    const float4 qv = *(const float4*)(Q + (size_t)row * LATDIM + lane * 4);
    const float4 kv = *(const float4*)(K + (size_t)col * LATDIM + lane * 4);
    float part = qv.x * kv.x + qv.y * kv.y + qv.z * kv.z + qv.w * kv.w;

    // reduce within the 8 lanes of each head
    part += __shfl_xor(part, 1, 32);
    part += __shfl_xor(part, 2, 32);
    part += __shfl_xor(part, 4, 32);

    if ((lane & 7) == 0) {
        const int h = lane >> 3;
        float att = fminf(fmaxf(part, -10.0f), 10.0f);
        float ea  = expf(att);
        expAtt[(size_t)e * NHEAD + h] = ea;
        atomicAdd(&attNorm[(size_t)row * NHEAD + h], ea);
    }
}

// ---------------------------------------------------------------------------
// Kernel 3: normalize attention + weighted scatter of V into resEmbeds.
// One wave per edge; lane l handles dims [4l, 4l+4), head = l/8.
// Normalizes att in place (expAtt region of d_out) and scatter-adds a*V[col].
// ---------------------------------------------------------------------------
__global__ __launch_bounds__(256)
void edge_scatter_kernel(const float* __restrict__ V,
                         const int*   __restrict__ rows,
                         const int*   __restrict__ cols,
                         const float* __restrict__ attNorm,  // [N, NHEAD]
                         float* __restrict__ att,            // [E, NHEAD] in/out
                         float* __restrict__ resEmbeds,      // [N, LATDIM]
                         int n_edges) {
    const int gid  = blockIdx.x * blockDim.x + threadIdx.x;
    const int e    = gid >> 5;
    const int lane = threadIdx.x & 31;
    if (e >= n_edges) return;

    const int row = rows[e];
    const int col = cols[e];
    const int h   = lane >> 3;

    const float ea   = att[(size_t)e * NHEAD + h];
    const float norm = attNorm[(size_t)row * NHEAD + h];
    const float a    = ea / (norm + 1e-8f);

    if ((lane & 7) == 0) att[(size_t)e * NHEAD + h] = a;

    const float4 vv = *(const float4*)(V + (size_t)col * LATDIM + lane * 4);
    float* dst = resEmbeds + (size_t)row * LATDIM + lane * 4;
    atomicAdd(dst + 0, a * vv.x);
    atomicAdd(dst + 1, a * vv.y);
    atomicAdd(dst + 2, a * vv.z);
    atomicAdd(dst + 3, a * vv.w);
}

// ---------------------------------------------------------------------------
extern "C" void kernel_launch(void* const* d_in, const int* in_sizes, int n_in,
                              void* d_out, int out_size, void* d_ws, size_t ws_size,
                              hipStream_t stream) {
    const float* embeds = (const float*)d_in[0];
    const float* qW     = (const float*)d_in[1];
    const float* kW     = (const float*)d_in[2];
    const float* vW     = (const float*)d_in[3];
    const int*   rows   = (const int*)d_in[4];
    const int*   cols   = (const int*)d_in[5];

    const int n_nodes = in_sizes[0] / LATDIM;   // 100000
    const int n_edges = in_sizes[4];            // 1600000

    // Workspace layout: Q | K | V | attNorm
    float* Qb      = (float*)d_ws;
    float* Kb      = Qb + (size_t)n_nodes * LATDIM;
    float* Vb      = Kb + (size_t)n_nodes * LATDIM;
    float* attNorm = Vb + (size_t)n_nodes * LATDIM;

    // Output layout: resEmbeds [N,128] then att [E,4]
    float* resEmbeds = (float*)d_out;
    float* attOut    = resEmbeds + (size_t)n_nodes * LATDIM;

    hipMemsetAsync(resEmbeds, 0, (size_t)n_nodes * LATDIM * sizeof(float), stream);
    hipMemsetAsync(attNorm,   0, (size_t)n_nodes * NHEAD  * sizeof(float), stream);

    // Phase 1: QKV projections (WMMA). 6250 row tiles x (3 mats * 8 col tiles).
    dim3 g1((n_nodes + 15) / 16, 24);
    qkv_wmma_kernel<<<g1, 32, 0, stream>>>(embeds, qW, kW, vW, Qb, Kb, Vb);

    // Phase 2/3: one wave per edge, 8 waves (256 threads) per block.
    const int edgesPerBlock = 8;
    const int blocks = (n_edges + edgesPerBlock - 1) / edgesPerBlock;
    edge_logits_kernel<<<blocks, 256, 0, stream>>>(Qb, Kb, rows, cols,
                                                   attOut, attNorm, n_edges);
    edge_scatter_kernel<<<blocks, 256, 0, stream>>>(Vb, rows, cols, attNorm,
                                                    attOut, resEmbeds, n_edges);
}